// GCN_84945863180627
// MI455X (gfx1250) — compile-verified
//
#include <hip/hip_runtime.h>

// ---------------- problem constants (from reference) ----------------
#define N_NODES   50000
#define N_EDGES   800000
#define N_GRAPHS  64
#define DF        96          // D_IN == D_HID == 96
#define D_OUT     2
#define NEG_SLOPE 0.2f
#define NTILES_N  (DF / 16)   // 6 output tiles per 16-row strip

typedef __attribute__((ext_vector_type(2))) float v2f;
typedef __attribute__((ext_vector_type(4))) float v4f;
typedef __attribute__((ext_vector_type(8))) float v8f;
typedef __attribute__((ext_vector_type(4))) int   v4i;

// address-space-qualified pointer types for the async global->LDS builtin
typedef v4i __attribute__((address_space(1)))* gptr_v4i;   // global
typedef v4i __attribute__((address_space(3)))* lptr_v4i;   // LDS

// ---------------- utility: zero a float buffer ----------------
__global__ void zero_f32(float* __restrict__ p, int n) {
  int i = blockIdx.x * blockDim.x + threadIdx.x;
  if (i < n) p[i] = 0.0f;
}

// ---------------- degree count via float atomics ----------------
__global__ void deg_count(const int* __restrict__ dst, float* __restrict__ deg) {
  int e = blockIdx.x * blockDim.x + threadIdx.x;
  if (e < N_EDGES) atomicAdd(&deg[dst[e]], 1.0f);
}

// deg -> dinv = rsqrt(deg + 1)   (in place)
__global__ void deg_to_dinv(float* __restrict__ deg) {
  int i = blockIdx.x * blockDim.x + threadIdx.x;
  if (i < N_NODES) deg[i] = rsqrtf(deg[i] + 1.0f);
}

// ---------------- fp32 WMMA GEMM: C[N_NODES,96] = A[N_NODES,96] @ W[96,96] ----------------
// Block = 256 threads (8 waves). W (36 KB) is staged once per block into LDS
// via gfx1250 async global->LDS (fallback: vectorized cooperative copy).
// Each wave computes one 16x96 strip: 6 accumulators (48 VGPRs), each A
// fragment reused across 6 WMMAs, B fragments served from LDS
// (conflict-free: stride 96 floats, lanes 0-15 consecutive).
__global__ void __launch_bounds__(256)
gemm_wmma_f32(const float* __restrict__ A,
              const float* __restrict__ W,
              float* __restrict__ C) {
  __shared__ float ldsW[DF * DF];   // 96x96 f32 = 36 KB

  const int tid = threadIdx.x;

  // ---- stage W into LDS ----
#if __has_builtin(__builtin_amdgcn_global_load_async_to_lds_b128)
  for (int i = tid; i < (DF * DF) / 4; i += 256) {
    __builtin_amdgcn_global_load_async_to_lds_b128(
        (gptr_v4i)(W + (size_t)i * 4),
        (lptr_v4i)(&ldsW[i * 4]),
        /*offset=*/0, /*cpol=*/0);
  }
  asm volatile("s_wait_asynccnt 0" ::: "memory");
#else
  for (int i = tid; i < (DF * DF) / 4; i += 256) {
    ((v4f*)ldsW)[i] = ((const v4f*)W)[i];
  }
#endif
  __syncthreads();

  const int wave  = tid >> 5;
  const int lane  = tid & 31;
  const int mTile = blockIdx.x * 8 + wave;
  if (mTile >= N_NODES / 16) return;      // uniform per-wave exit; no later barriers

  const int row0 = mTile * 16;
  const int half = lane >> 4;             // 0: lanes 0-15, 1: lanes 16-31
  const int l16  = lane & 15;

  v8f acc[NTILES_N] = {};
  // A-frag per ISA 7.12.2: lanes 0-15 hold K={k,k+1}, lanes 16-31 K={k+2,k+3}
  const float* arow = A + (size_t)(row0 + l16) * DF + 2 * half;

  for (int k = 0; k < DF; k += 4) {
    v2f a;
    a.x = arow[k + 0];
    a.y = arow[k + 1];
    const float* wrow = ldsW + (k + 2 * half) * DF + l16;
    #pragma unroll
    for (int t = 0; t < NTILES_N; ++t) {
      v2f b;
      b.x = wrow[t * 16];
      b.y = wrow[t * 16 + DF];
      // 8 args: (neg_a, A, neg_b, B, c_mod, C, reuse_a, reuse_b)
      acc[t] = __builtin_amdgcn_wmma_f32_16x16x4_f32(false, a, false, b,
                                                     (short)0, acc[t],
                                                     false, false);
    }
  }

  // C/D layout: VGPR r -> M = r + 8*half, N = l16 (per tile)
  #pragma unroll
  for (int t = 0; t < NTILES_N; ++t) {
    #pragma unroll
    for (int r = 0; r < 8; ++r) {
      C[(size_t)(row0 + r + 8 * half) * DF + t * 16 + l16] = acc[t][r];
    }
  }
}

// ---------------- edge aggregation: agg[dst] += hW[src] * dinv[src]*dinv[dst] ----------------
// One wave per edge; lane f handles features f, f+32, f+64 (coalesced atomics).
__global__ void edge_agg(const int* __restrict__ src, const int* __restrict__ dst,
                         const float* __restrict__ hW, const float* __restrict__ dinv,
                         float* __restrict__ agg) {
  int wave = (blockIdx.x * blockDim.x + threadIdx.x) >> 5;
  int lane = threadIdx.x & 31;
  if (wave >= N_EDGES) return;
  int s = src[wave];
  int d = dst[wave];
  float coef = dinv[s] * dinv[d];
  const float* hs = hW + (size_t)s * DF;
  float*       ad = agg + (size_t)d * DF;
  #pragma unroll
  for (int f = lane; f < DF; f += 32) {
    atomicAdd(&ad[f], hs[f] * coef);
  }
}

// ---------------- out = leaky(agg + hW * dinv^2 + bias) ----------------
__global__ void combine_leaky(const float* __restrict__ hW, const float* __restrict__ agg,
                              const float* __restrict__ dinv, const float* __restrict__ bias,
                              float* __restrict__ out) {
  int i = blockIdx.x * blockDim.x + threadIdx.x;
  if (i >= N_NODES * DF) return;
  int node = i / DF;
  int f    = i - node * DF;
  float di = dinv[node];
  float v  = agg[i] + hW[i] * (di * di) + bias[f];
  out[i] = (v > 0.0f) ? v : NEG_SLOPE * v;
}

// ---------------- pooled[batch[n]] += h[n]; cnt[batch[n]] += 1 ----------------
__global__ void pool_acc(const float* __restrict__ h, const int* __restrict__ batch,
                         float* __restrict__ pooled, float* __restrict__ cnt) {
  int wave = (blockIdx.x * blockDim.x + threadIdx.x) >> 5;
  int lane = threadIdx.x & 31;
  if (wave >= N_NODES) return;
  int g = batch[wave];
  const float* hn = h + (size_t)wave * DF;
  float*       pg = pooled + (size_t)g * DF;
  #pragma unroll
  for (int f = lane; f < DF; f += 32) {
    atomicAdd(&pg[f], hn[f]);
  }
  if (lane == 0) atomicAdd(&cnt[g], 1.0f);
}

// ---------------- out[g,o] = (pooled[g]/max(cnt,1)) @ Wc + bc ----------------
__global__ void classify(const float* __restrict__ pooled, const float* __restrict__ cnt,
                         const float* __restrict__ Wc, const float* __restrict__ bc,
                         float* __restrict__ out) {
  int i = blockIdx.x * blockDim.x + threadIdx.x;
  if (i >= N_GRAPHS * D_OUT) return;
  int g = i / D_OUT;
  int o = i - g * D_OUT;
  float inv = 1.0f / fmaxf(cnt[g], 1.0f);
  const float* pg = pooled + (size_t)g * DF;
  float s = 0.0f;
  #pragma unroll 8
  for (int f = 0; f < DF; ++f) s = fmaf(pg[f], Wc[f * D_OUT + o], s);
  out[i] = s * inv + bc[o];
}

// ---------------- host-side launch ----------------
extern "C" void kernel_launch(void* const* d_in, const int* in_sizes, int n_in,
                              void* d_out, int out_size, void* d_ws, size_t ws_size,
                              hipStream_t stream) {
  (void)in_sizes; (void)n_in; (void)out_size; (void)ws_size;

  const float* x     = (const float*)d_in[0];
  const int*   eidx  = (const int*)d_in[1];     // [2, N_EDGES]: row0 = src, row1 = dst
  const int*   batch = (const int*)d_in[2];
  const float* W1    = (const float*)d_in[3];
  const float* b1    = (const float*)d_in[4];
  const float* W2    = (const float*)d_in[5];
  const float* b2    = (const float*)d_in[6];
  const float* Wc    = (const float*)d_in[7];
  const float* bc    = (const float*)d_in[8];
  const int* src = eidx;
  const int* dst = eidx + N_EDGES;
  float* out = (float*)d_out;

  // workspace layout (floats)
  float* ws   = (float*)d_ws;
  float* dinv = ws;                                 // N_NODES
  float* bufA = dinv + N_NODES;                     // N_NODES * DF  (hW)
  float* bufB = bufA + (size_t)N_NODES * DF;        // N_NODES * DF  (agg)
  float* bufC = bufB + (size_t)N_NODES * DF;        // N_NODES * DF  (layer out)
  float* pooled = bufC + (size_t)N_NODES * DF;      // N_GRAPHS * DF
  float* cnt    = pooled + (size_t)N_GRAPHS * DF;   // N_GRAPHS

  const int NF = N_NODES * DF;                      // 4,800,000

  // ---- degree / dinv (shared by both layers) ----
  zero_f32<<<(N_NODES + 255) / 256, 256, 0, stream>>>(dinv, N_NODES);
  deg_count<<<(N_EDGES + 255) / 256, 256, 0, stream>>>(dst, dinv);
  deg_to_dinv<<<(N_NODES + 255) / 256, 256, 0, stream>>>(dinv);

  const int mTiles     = N_NODES / 16;              // 3125
  const int gemmBlocks = (mTiles + 7) / 8;          // 8 strip-waves per block
  const int edgeBlocks = N_EDGES / 8;               // 8 edge-waves per block
  const int nodeWaveBlocks = (N_NODES + 7) / 8;     // 8 node-waves per block

  // ---- layer 1 ----
  gemm_wmma_f32<<<gemmBlocks, 256, 0, stream>>>(x, W1, bufA);
  zero_f32<<<(NF + 255) / 256, 256, 0, stream>>>(bufB, NF);
  edge_agg<<<edgeBlocks, 256, 0, stream>>>(src, dst, bufA, dinv, bufB);
  combine_leaky<<<(NF + 255) / 256, 256, 0, stream>>>(bufA, bufB, dinv, b1, bufC);

  // ---- layer 2 ----
  gemm_wmma_f32<<<gemmBlocks, 256, 0, stream>>>(bufC, W2, bufA);
  zero_f32<<<(NF + 255) / 256, 256, 0, stream>>>(bufB, NF);
  edge_agg<<<edgeBlocks, 256, 0, stream>>>(src, dst, bufA, dinv, bufB);
  combine_leaky<<<(NF + 255) / 256, 256, 0, stream>>>(bufA, bufB, dinv, b2, bufC);

  // ---- mean pool + classifier ----
  zero_f32<<<(N_GRAPHS * DF + N_GRAPHS + 255) / 256, 256, 0, stream>>>(pooled, N_GRAPHS * DF + N_GRAPHS);
  pool_acc<<<nodeWaveBlocks, 256, 0, stream>>>(bufC, batch, pooled, cnt);
  classify<<<(N_GRAPHS * D_OUT + 127) / 128, 128, 0, stream>>>(pooled, cnt, Wc, bc, out);
}